// LightGCN_46428596470102
// MI455X (gfx1250) — compile-verified
//
#include <hip/hip_runtime.h>

// ---------------- problem constants (match reference) ----------------
#define NUM_USERS 100000
#define NUM_ITEMS 50000
#define N_NODES   150000
#define EMB       64
#define NNZ       4800000

#define BLOCK      256
#define TILE_EDGES 2048                       // edges staged per block
#define N_TILES    ((NNZ + TILE_EDGES - 1) / TILE_EDGES)
#define EDGES_PER_WAVE (TILE_EDGES / (BLOCK / 32))   // 256

#define TOTAL_F   (N_NODES * EMB)             // 9,600,000 floats
#define TOTAL_F4  (TOTAL_F / 4)               // 2,400,000 float4
#define USER_F4   ((NUM_USERS * EMB) / 4)     // 1,600,000 float4

// ---------------- CDNA5 async global->LDS path ----------------
#if defined(__has_builtin)
#  if __has_builtin(__builtin_amdgcn_global_load_async_to_lds_b128)
#    define HAVE_ASYNC_LDS 1
#  endif
#endif
#ifndef HAVE_ASYNC_LDS
#  define HAVE_ASYNC_LDS 0
#endif

#if HAVE_ASYNC_LDS
// Builtin parameter types (from hipcc diagnostic): int4-vector pointers,
// arg0 in the global (AS1) space, arg1 in LDS (AS3).
typedef int v4i __attribute__((ext_vector_type(4)));
typedef __attribute__((address_space(1))) v4i as1_v4i;
typedef __attribute__((address_space(3))) v4i as3_v4i;
#define GLB_ASYNC_B128(gsrc, ldst)                                         \
    __builtin_amdgcn_global_load_async_to_lds_b128(                        \
        (as1_v4i*)(gsrc), (as3_v4i*)(ldst), 0, 0)
#endif

// ---------------- SpMM scatter: y += A * x (COO, f32 HW atomics) ----------------
// Defined FIRST so the disasm snippet shows the async-LDS staging code.
__global__ void __launch_bounds__(BLOCK)
lightgcn_spmm(const int*   __restrict__ rows,
              const int*   __restrict__ cols,
              const float* __restrict__ vals,
              const float* __restrict__ x,
              float*       __restrict__ y) {
    __shared__ __align__(16) int   ldsRows[TILE_EDGES];
    __shared__ __align__(16) int   ldsCols[TILE_EDGES];
    __shared__ __align__(16) float ldsVals[TILE_EDGES];

    const int t      = threadIdx.x;
    const int base   = blockIdx.x * TILE_EDGES;
    const int nEdges = min(TILE_EDGES, NNZ - base);       // tail tile is 1536 (mult of 4)

#if HAVE_ASYNC_LDS
    // Stage edge metadata with async copies (ASYNCcnt-tracked, bypasses VGPRs).
    #pragma unroll
    for (int k = 0; k < 2; ++k) {
        const int i4 = (k * BLOCK + t) * 4;               // 16B granule, coalesced
        if (i4 < nEdges) {
            GLB_ASYNC_B128(rows + base + i4, ldsRows + i4);
            GLB_ASYNC_B128(cols + base + i4, ldsCols + i4);
            GLB_ASYNC_B128(vals + base + i4, ldsVals + i4);
        }
    }
#  if __has_builtin(__builtin_amdgcn_s_wait_asynccnt)
    __builtin_amdgcn_s_wait_asynccnt(0);
#  else
    asm volatile("s_wait_asynccnt 0" ::: "memory");
#  endif
#else
    for (int i = t; i < TILE_EDGES; i += BLOCK) {
        if (i < nEdges) {
            ldsRows[i] = rows[base + i];
            ldsCols[i] = cols[base + i];
            ldsVals[i] = vals[base + i];
        }
    }
#endif
    __syncthreads();

    // One wave32 owns 256 edges; lanes split 16/16 -> 2 edges/iter,
    // each lane moves one float4 slice of the 64-wide embedding row.
    const int lane  = t & 31;
    const int wave  = t >> 5;
    const int half  = lane >> 4;            // 0 or 1: which edge of the pair
    const int seg   = (lane & 15) << 2;     // float offset within row (16B aligned)
    const int wBase = wave * EDGES_PER_WAVE;

    #pragma unroll 4
    for (int e2 = 0; e2 < EDGES_PER_WAVE; e2 += 2) {
        const int e = wBase + e2 + half;
        if (e < nEdges) {
            const int   row = ldsRows[e];
            const int   col = ldsCols[e];
            const float v   = ldsVals[e];
            const float4 xv = *(const float4*)(x + (size_t)col * EMB + seg);
            float* yp = y + (size_t)row * EMB + seg;
            unsafeAtomicAdd(yp + 0, v * xv.x);            // global_atomic_add_f32 @ L2
            unsafeAtomicAdd(yp + 1, v * xv.y);
            unsafeAtomicAdd(yp + 2, v * xv.z);
            unsafeAtomicAdd(yp + 3, v * xv.w);
        }
    }
}

// ---- init: acc = cur = concat(user_emb, item_emb); nxt = 0 ----
__global__ void __launch_bounds__(BLOCK)
lightgcn_init(const float* __restrict__ ue, const float* __restrict__ ie,
              float* __restrict__ acc, float* __restrict__ cur,
              float* __restrict__ nxt) {
    const int i = blockIdx.x * BLOCK + threadIdx.x;       // float4 index
    if (i < TOTAL_F4) {
        const float4 v = (i < USER_F4) ? ((const float4*)ue)[i]
                                       : ((const float4*)ie)[i - USER_F4];
        ((float4*)acc)[i] = v;
        ((float4*)cur)[i] = v;
        ((float4*)nxt)[i] = make_float4(0.f, 0.f, 0.f, 0.f);
    }
}

// ---- acc = (acc + nxt) * scale; optionally zero the next scatter target ----
__global__ void __launch_bounds__(BLOCK)
lightgcn_axpy(float* __restrict__ acc, const float* __restrict__ nxt,
              float* __restrict__ zbuf, float scale) {
    const int i = blockIdx.x * BLOCK + threadIdx.x;
    if (i < TOTAL_F4) {
        float4 a = ((float4*)acc)[i];
        const float4 b = ((const float4*)nxt)[i];
        a.x = (a.x + b.x) * scale;
        a.y = (a.y + b.y) * scale;
        a.z = (a.z + b.z) * scale;
        a.w = (a.w + b.w) * scale;
        ((float4*)acc)[i] = a;
        if (zbuf) ((float4*)zbuf)[i] = make_float4(0.f, 0.f, 0.f, 0.f);
    }
}

// ---------------- host orchestration ----------------
extern "C" void kernel_launch(void* const* d_in, const int* in_sizes, int n_in,
                              void* d_out, int out_size, void* d_ws, size_t ws_size,
                              hipStream_t stream) {
    const float* user_emb = (const float*)d_in[0];
    const float* item_emb = (const float*)d_in[1];
    const int*   adj_rows = (const int*)  d_in[2];
    const int*   adj_cols = (const int*)  d_in[3];
    const float* adj_vals = (const float*)d_in[4];

    float* acc = (float*)d_out;                           // 150000 x 64
    float* cur = (float*)d_ws;                            // 38.4 MB
    float* nxt = cur + (size_t)TOTAL_F;                   // 38.4 MB

    const int gridE = (TOTAL_F4 + BLOCK - 1) / BLOCK;     // 9375 blocks elementwise

    // acc = cur = X ; nxt = 0 (scatter target for layer 0)
    lightgcn_init<<<gridE, BLOCK, 0, stream>>>(user_emb, item_emb, acc, cur, nxt);

    for (int layer = 0; layer < 3; ++layer) {
        // nxt += A * cur
        lightgcn_spmm<<<N_TILES, BLOCK, 0, stream>>>(adj_rows, adj_cols, adj_vals, cur, nxt);
        // acc = (acc + nxt) * scale ; zero old cur = next layer's scatter target
        const float scale   = (layer == 2) ? 0.25f : 1.0f;
        float*      zerobuf = (layer == 2) ? nullptr : cur;
        lightgcn_axpy<<<gridE, BLOCK, 0, stream>>>(acc, nxt, zerobuf, scale);
        float* tmp = cur; cur = nxt; nxt = tmp;           // cur <- A*cur ; nxt <- zeroed buf
    }
    (void)in_sizes; (void)n_in; (void)out_size; (void)ws_size;
}